// AgentRuleGNN_19061064860173
// MI455X (gfx1250) — compile-verified
//
#include <hip/hip_runtime.h>

typedef __attribute__((ext_vector_type(2))) float v2f;
typedef __attribute__((ext_vector_type(8))) float v8f;
typedef __attribute__((ext_vector_type(4))) int   i4;

// address-space-qualified vector pointees for the gfx1250 async-to-LDS builtin
typedef i4 __attribute__((address_space(1))) as1_i4;   // global
typedef i4 __attribute__((address_space(3))) as3_i4;   // LDS

#define N_AGENTS 2048
#define D 8     // STATE_DIM
#define H 32    // HIDDEN_DIM
#define M 16    // MSG_DIM
#define TJ 64   // sender tile rows staged in LDS

#if defined(__gfx1250__) &&                                                    \
    __has_builtin(__builtin_amdgcn_global_load_async_to_lds_b128) &&           \
    __has_builtin(__builtin_amdgcn_s_wait_asynccnt)
#define USE_ASYNC 1
#else
#define USE_ASYNC 0
#endif

// ---- f32 WMMA 16x16x4 fragment helpers -------------------------------------
// A (16x4, MxK): lane m (0..15) holds A[m,k0],A[m,k0+1]; lane m+16 holds
// A[m,k0+2],A[m,k0+3].
static __device__ __forceinline__ v2f ldA(const float* A, int lda, int k0, int lane) {
    const int m = lane & 15, hi = lane >> 4;
    const float* p = A + m * lda + k0 + 2 * hi;
    v2f a; a.x = p[0]; a.y = p[1];
    return a;
}
// B (4x16, KxN): VGPR0 = rows k0 (lanes 0-15) / k0+2 (lanes 16-31),
//                VGPR1 = rows k0+1 / k0+3; column = lane&15.
static __device__ __forceinline__ v2f ldB(const float* B, int ldb, int k0, int lane) {
    const int n = lane & 15, hi = lane >> 4;
    v2f b;
    b.x = B[(k0 + 2 * hi) * ldb + n];
    b.y = B[(k0 + 2 * hi + 1) * ldb + n];
    return b;
}
static __device__ __forceinline__ v8f wmma4(v2f a, v2f b, v8f c) {
    return __builtin_amdgcn_wmma_f32_16x16x4_f32(false, a, false, b, (short)0, c,
                                                 false, false);
}

// ---- async (or fallback) copy of one TJ x H sender tile into LDS -----------
// 8 KB tile / 256 threads / 16 B = 2 b128 transfers per thread (ASYNC_PER_TILE).
#define ASYNC_PER_TILE 2
static __device__ __forceinline__ void copy_tile(float* dst, const float* src,
                                                 int tid) {
#if USE_ASYNC
    for (int t = tid; t < TJ * H / 4; t += 256) {
        __builtin_amdgcn_global_load_async_to_lds_b128(
            (as1_i4*)(src + t * 4), (as3_i4*)(dst + t * 4), 0, 0);
    }
#else
    for (int t = tid; t < TJ * H / 4; t += 256)
        ((float4*)dst)[t] = ((const float4*)src)[t];
#endif
}
template <int N>
static __device__ __forceinline__ void wait_tiles() {
#if USE_ASYNC
    __builtin_amdgcn_s_wait_asynccnt(N);
#endif
}

// ---- Kernel 1: SL = states @ W1[:D]; RB = states @ W1[D:] + b1 -------------
// grid (128, 4), block 32. blockIdx.y: 0,1 -> SL col halves; 2,3 -> RB halves.
__global__ void __launch_bounds__(32)
lin_kernel(const float* __restrict__ states, const float* __restrict__ W1,
           const float* __restrict__ b1, float* __restrict__ SL,
           float* __restrict__ RB) {
    const int lane = threadIdx.x;
    const int row0 = blockIdx.x * 16;
    const int which = blockIdx.y >> 1;   // 0 = sender half of W1, 1 = receiver
    const int sub   = blockIdx.y & 1;    // which 16-col half of the 32 outputs
    const int n = lane & 15, hi = lane >> 4;

    const float* A = states + row0 * D;
    const float* B = W1 + which * D * H + sub * 16;

    v8f c;
#pragma unroll
    for (int r = 0; r < 8; ++r) c[r] = which ? b1[sub * 16 + n] : 0.0f;

#pragma unroll
    for (int k0 = 0; k0 < D; k0 += 4)
        c = wmma4(ldA(A, D, k0, lane), ldB(B, H, k0, lane), c);

    float* out = (which ? RB : SL) + row0 * H + sub * 16;
#pragma unroll
    for (int r = 0; r < 8; ++r) out[(r + 8 * hi) * H + n] = c[r];
}

// ---- Kernel 2: masked relu aggregation over all sender/receiver pairs ------
// AGGN[i,k] = (sum_j adj[i,j] * relu(RB[i,k] + SL[j,k])) / max(deg,1)
// Two receivers per wave (each staged sender value feeds two fma chains),
// sender tiles double-buffered in LDS via gfx1250 async-to-LDS loads,
// adjacency streamed with non-temporal b128 loads (read exactly once).
__global__ void __launch_bounds__(256)
pair_agg_kernel(const int* __restrict__ adj, const float* __restrict__ SL,
                const float* __restrict__ RB, float* __restrict__ AGGN,
                float* __restrict__ BSC) {
    __shared__ float slbuf[2][TJ * H];   // 2 x 8 KB sender tiles
    const int tid  = threadIdx.x;
    const int lane = tid & 31;
    const int wave = tid >> 5;
    const int i0 = blockIdx.x * 16 + wave * 2;
    const int i1 = i0 + 1;

    const float rb0 = RB[i0 * H + lane];
    const float rb1 = RB[i1 * H + lane];
    const int* adjrow0 = adj + (long)i0 * N_AGENTS;
    const int* adjrow1 = adj + (long)i1 * N_AGENTS;
    float acc0 = 0.0f, acc1 = 0.0f;
    int deg0 = 0, deg1 = 0;

    copy_tile(&slbuf[0][0], SL, tid);    // prefetch tile 0

    for (int jb = 0; jb < N_AGENTS; jb += TJ) {
        const int buf = (jb / TJ) & 1;
        const bool pref = (jb + TJ) < N_AGENTS;
        __syncthreads();                 // all reads of slbuf[buf^1] retired
        if (pref) {
            copy_tile(&slbuf[buf ^ 1][0], SL + (jb + TJ) * H, tid);
            wait_tiles<ASYNC_PER_TILE>();   // tile jb landed; next still in flight
        } else {
            wait_tiles<0>();                // last tile: drain everything
        }
        __syncthreads();                 // tile jb visible to every wave

        const float* sb = &slbuf[buf][0];
#pragma unroll 4
        for (int jj = 0; jj < TJ; jj += 4) {
            const i4 a0 = __builtin_nontemporal_load((const i4*)(adjrow0 + jb + jj));
            const i4 a1 = __builtin_nontemporal_load((const i4*)(adjrow1 + jb + jj));
#pragma unroll
            for (int t = 0; t < 4; ++t) {
                const float sl = sb[(jj + t) * H + lane];
                const float h0 = fmaxf(rb0 + sl, 0.0f);
                const float h1 = fmaxf(rb1 + sl, 0.0f);
                acc0 = fmaf((float)a0[t], h0, acc0);
                acc1 = fmaf((float)a1[t], h1, acc1);
                deg0 += a0[t];
                deg1 += a1[t];
            }
        }
    }

    AGGN[i0 * H + lane] = acc0 / fmaxf((float)deg0, 1.0f);
    AGGN[i1 * H + lane] = acc1 / fmaxf((float)deg1, 1.0f);
    if (lane == 0) {
        BSC[i0] = (deg0 > 0) ? 1.0f : 0.0f;
        BSC[i1] = (deg1 > 0) ? 1.0f : 0.0f;
    }
}

// ---- Kernel 3: fused epilogue, one wave per 16 agents ----------------------
// aggregated = AGGN @ W2 + b2*bsc ; u = [states | aggregated]
// u1 = relu(u @ U1 + c1) ; u2 = relu(u1 @ U2 + c2) ; out = u2 @ U3 + c3
__global__ void __launch_bounds__(32)
update_kernel(const float* __restrict__ states, const float* __restrict__ AGGN,
              const float* __restrict__ BSC, const float* __restrict__ W2,
              const float* __restrict__ b2, const float* __restrict__ U1,
              const float* __restrict__ c1, const float* __restrict__ U2,
              const float* __restrict__ c2, const float* __restrict__ U3,
              const float* __restrict__ c3, float* __restrict__ out) {
    __shared__ float ub[16 * 25];    // u tile  [16 x 24], stride 25
    __shared__ float u1b[16 * 33];   // u1 tile [16 x 32], stride 33
    __shared__ float u2b[16 * 33];   // u2 tile [16 x 32], stride 33

    const int lane = threadIdx.x;
    const int row0 = blockIdx.x * 16;
    const int n = lane & 15, hi = lane >> 4;

    // aggregated = AGGN_tile @ W2 + b2 * bscale  (K=32, N=16)
    v8f c;
#pragma unroll
    for (int r = 0; r < 8; ++r) c[r] = b2[n] * BSC[row0 + r + 8 * hi];
    const float* A0 = AGGN + row0 * H;
#pragma unroll
    for (int k0 = 0; k0 < H; k0 += 4)
        c = wmma4(ldA(A0, H, k0, lane), ldB(W2, M, k0, lane), c);

    // stage u = [states | aggregated] into LDS
    for (int idx = lane; idx < 16 * D; idx += 32) {
        const int r = idx >> 3, cc = idx & 7;
        ub[r * 25 + cc] = states[(row0 + r) * D + cc];
    }
#pragma unroll
    for (int r = 0; r < 8; ++r) ub[(r + 8 * hi) * 25 + 8 + n] = c[r];
    __syncthreads();

    // u1 = relu(u @ U1 + c1)   (K=24, N=32 -> two 16-col tiles)
#pragma unroll
    for (int nt = 0; nt < 2; ++nt) {
        v8f cc1;
#pragma unroll
        for (int r = 0; r < 8; ++r) cc1[r] = c1[nt * 16 + n];
#pragma unroll
        for (int k0 = 0; k0 < 24; k0 += 4)
            cc1 = wmma4(ldA(ub, 25, k0, lane), ldB(U1 + nt * 16, H, k0, lane), cc1);
#pragma unroll
        for (int r = 0; r < 8; ++r)
            u1b[(r + 8 * hi) * 33 + nt * 16 + n] = fmaxf(cc1[r], 0.0f);
    }
    __syncthreads();

    // u2 = relu(u1 @ U2 + c2)  (K=32, N=32)
#pragma unroll
    for (int nt = 0; nt < 2; ++nt) {
        v8f cc2;
#pragma unroll
        for (int r = 0; r < 8; ++r) cc2[r] = c2[nt * 16 + n];
#pragma unroll
        for (int k0 = 0; k0 < H; k0 += 4)
            cc2 = wmma4(ldA(u1b, 33, k0, lane), ldB(U2 + nt * 16, H, k0, lane), cc2);
#pragma unroll
        for (int r = 0; r < 8; ++r)
            u2b[(r + 8 * hi) * 33 + nt * 16 + n] = fmaxf(cc2[r], 0.0f);
    }
    __syncthreads();

    // out = u2 @ U3 + c3   (K=32, N=8 zero-padded to 16)
    v8f c3t;
#pragma unroll
    for (int r = 0; r < 8; ++r) c3t[r] = (n < D) ? c3[n] : 0.0f;
#pragma unroll
    for (int k0 = 0; k0 < H; k0 += 4) {
        v2f b; b.x = 0.0f; b.y = 0.0f;
        if (n < D) {
            b.x = U3[(k0 + 2 * hi) * D + n];
            b.y = U3[(k0 + 2 * hi + 1) * D + n];
        }
        c3t = wmma4(ldA(u2b, 33, k0, lane), b, c3t);
    }
    if (n < D) {
#pragma unroll
        for (int r = 0; r < 8; ++r) out[(row0 + r + 8 * hi) * D + n] = c3t[r];
    }
}

extern "C" void kernel_launch(void* const* d_in, const int* in_sizes, int n_in,
                              void* d_out, int out_size, void* d_ws, size_t ws_size,
                              hipStream_t stream) {
    const float* states = (const float*)d_in[0];
    const int*   adj    = (const int*)d_in[1];
    const float* W1     = (const float*)d_in[2];
    const float* b1     = (const float*)d_in[3];
    const float* W2     = (const float*)d_in[4];
    const float* b2     = (const float*)d_in[5];
    const float* U1     = (const float*)d_in[6];
    const float* c1     = (const float*)d_in[7];
    const float* U2     = (const float*)d_in[8];
    const float* c2     = (const float*)d_in[9];
    const float* U3     = (const float*)d_in[10];
    const float* c3     = (const float*)d_in[11];
    float* out = (float*)d_out;

    // workspace layout (floats): SL[2048*32] RB[2048*32] AGGN[2048*32] BSC[2048]
    float* SL   = (float*)d_ws;
    float* RB   = SL + N_AGENTS * H;
    float* AGGN = RB + N_AGENTS * H;
    float* BSC  = AGGN + N_AGENTS * H;

    lin_kernel<<<dim3(N_AGENTS / 16, 4), 32, 0, stream>>>(states, W1, b1, SL, RB);
    pair_agg_kernel<<<N_AGENTS / 16, 256, 0, stream>>>(adj, SL, RB, AGGN, BSC);
    update_kernel<<<N_AGENTS / 16, 32, 0, stream>>>(states, AGGN, BSC, W2, b2, U1,
                                                    c1, U2, c2, U3, c3, out);
}